// BiRNNLM_67147518706378
// MI455X (gfx1250) — compile-verified
//
#include <hip/hip_runtime.h>
#include <hip/hip_bf16.h>
#include <math.h>

// ---------------------------------------------------------------------------
// BiRNN LM:  S=128, B=32, VOCAB=32000, EMB=32, HID=16
// d_in: 0=input_batch(int32 [S,B]) 1=h0_lr[B,H] 2=h0_rl[B,H] 3=embedding[V,E]
//       4=W_lr[H,E+H] 5=b_lr[H] 6=W_rl[H,E+H] 7=b_rl[H] 8=W_ho[V,2H] 9=b_ho[V]
// d_out: log_softmax logits [S,B,V] f32
// ---------------------------------------------------------------------------

#define S_LEN 128
#define BATCH 32
#define VOCAB 32000
#define EMB 32
#define HID 16
#define CDIM 32          // 2*HID, combined feature dim == GEMM K
#define NROWS (S_LEN*BATCH)          // 4096 GEMM M rows
#define NMTILE (NROWS/16)            // 256 M-tiles
#define NSPLIT 16                    // vocab splits per M-tile
#define TILES_PER_SPLIT (VOCAB/16/NSPLIT)   // 125

typedef __attribute__((ext_vector_type(2))) float v2f;
typedef __attribute__((ext_vector_type(8))) float v8f;

// Branchless single-exp online-softmax update: (m,s) <- merge with value x.
__device__ __forceinline__ void online_upd(float& m, float& s, float x)
{
    const float nm = fmaxf(m, x);
    const float e  = __expf(fminf(m, x) - nm);   // exp of the smaller - max
    const bool xbig = (x >= m);
    // if x is the new max: s = s*e + 1;  else: s = s + e
    s = fmaf(s, xbig ? e : 1.0f, xbig ? 1.0f : e);
    m = nm;
}

// Branchless single-exp merge of two (max,sum) pairs.
__device__ __forceinline__ void online_merge(float& m, float& s, float om, float os)
{
    const float nm = fmaxf(m, om);
    const float e  = __expf(fminf(m, om) - nm);
    const bool keep = (m >= om);                 // our max wins
    s = fmaf(s, keep ? 1.0f : e, os * (keep ? e : 1.0f));
    m = nm;
}

// ---------------------------------------------------------------------------
// Kernel 1: embedding gather + both RNN scans in one block.
// Threads 0..511: LR direction;  512..1023: RL direction.
// Within a half: tid = b*16 + j  (b = batch, j = hidden unit).
// Writes combined[S][B][32]: [:,:,0:16]=hLR, [:,:,16:32]=g where g[i]=hRL[S-1-i]
// ---------------------------------------------------------------------------
__global__ __launch_bounds__(1024) void birnn_scan_kernel(
    const int* __restrict__ tok,      // [S,B]
    const float* __restrict__ h0lr,   // [B,H]
    const float* __restrict__ h0rl,   // [B,H]
    const float* __restrict__ emb,    // [V,E]
    const float* __restrict__ Wlr,    // [H, E+H]
    const float* __restrict__ blr,    // [H]
    const float* __restrict__ Wrl,    // [H, E+H]
    const float* __restrict__ brl,    // [H]
    float* __restrict__ C)            // [S,B,32]
{
    __shared__ float xs[2][BATCH][EMB];
    __shared__ float hs[2][BATCH][HID];

    const int tid = threadIdx.x;
    const int dir = tid >> 9;          // 0 = LR, 1 = RL
    const int r   = tid & 511;
    const int b   = r >> 4;
    const int j   = r & 15;

    const float* W  = dir ? Wrl : Wlr;
    const float* bb = dir ? brl : blr;
    const float* h0 = dir ? h0rl : h0lr;

    // RNN weight row in registers (48 floats) + bias
    float w[EMB + HID];
#pragma unroll
    for (int k = 0; k < EMB + HID; ++k) w[k] = W[j * (EMB + HID) + k];
    const float bias = bb[j];

    // initial state: hLR[0] and g[S-1] = h0
    float h = h0[b * HID + j];
    hs[dir][b][j] = h;
    {
        const int i0 = dir ? (S_LEN - 1) : 0;
        C[(i0 * BATCH + b) * CDIM + dir * HID + j] = h;
    }
    __syncthreads();

    for (int step = 1; step < S_LEN; ++step) {
        const int i     = dir ? (S_LEN - 1 - step) : step;   // output index
        const int emb_t = dir ? (i + 1) : (step - 1);        // emb timestep

        // gather x = embedding[token]  (16 threads per batch row, 2 floats each)
        const int token = tok[emb_t * BATCH + b];
        xs[dir][b][j]        = emb[token * EMB + j];
        xs[dir][b][j + HID]  = emb[token * EMB + j + HID];
        __syncthreads();

        float acc = bias;
#pragma unroll
        for (int k = 0; k < EMB; ++k) acc = fmaf(w[k], xs[dir][b][k], acc);
#pragma unroll
        for (int k = 0; k < HID; ++k) acc = fmaf(w[EMB + k], hs[dir][b][k], acc);
        const float hn = tanhf(acc);
        __syncthreads();

        hs[dir][b][j] = hn;
        C[(i * BATCH + b) * CDIM + dir * HID + j] = hn;
        __syncthreads();
    }
}

// ---------------------------------------------------------------------------
// WMMA fragment loaders.
// f32 16x16x4 operand layout (ISA 7.12.2): lane L holds row (L%16),
// K-pair 2*(L/16)+{0,1} in its 2 VGPRs -> aligned float2 loads, both for
// A (combined, stride 32) and B (W_ho, stride 32).
// C layout: VGPR r, lane L -> M = r + 8*(L/16), N = L%16.
// ---------------------------------------------------------------------------
__device__ __forceinline__ void load_frag8(const float* __restrict__ base_row,
                                           v2f frag[8])
{
#pragma unroll
    for (int t = 0; t < 8; ++t)
        frag[t] = *reinterpret_cast<const v2f*>(base_row + 4 * t);
}

__device__ __forceinline__ v8f gemm_tile(const v2f a[8], const v2f bv[8])
{
    v8f c = {};
#pragma unroll
    for (int t = 0; t < 8; ++t) {
        c = __builtin_amdgcn_wmma_f32_16x16x4_f32(
                /*neg_a=*/false, a[t], /*neg_b=*/false, bv[t],
                /*c_mod=*/(short)0, c, /*reuse_a=*/false, /*reuse_b=*/false);
    }
    return c;
}

// ---------------------------------------------------------------------------
// Kernel 2: per-row online softmax stats over a vocab split, GEMM via WMMA.
// One wave per (mtile, split). Per-lane online (max,sum) with ONE exp per
// element, merged across the 16 lanes of each half-wave via shuffles.
// ---------------------------------------------------------------------------
__global__ __launch_bounds__(256) void logits_stats_kernel(
    const float* __restrict__ C,     // [4096,32]
    const float* __restrict__ Who,   // [V,32]
    const float* __restrict__ bho,   // [V]
    float* __restrict__ pMax,        // [4096,NSPLIT]
    float* __restrict__ pSum)        // [4096,NSPLIT]
{
    const int wave  = blockIdx.x * (blockDim.x >> 5) + (threadIdx.x >> 5);
    const int lane  = threadIdx.x & 31;
    const int mtile = wave / NSPLIT;
    const int split = wave % NSPLIT;
    const int m0    = mtile * 16;
    const int ln    = lane & 15;
    const int hi    = lane >> 4;
    const int ko    = hi * 2;

    v2f a[8];
    load_frag8(C + (m0 + ln) * CDIM + ko, a);

    float rmax[8], rsum[8];
#pragma unroll
    for (int r = 0; r < 8; ++r) { rmax[r] = -INFINITY; rsum[r] = 0.0f; }

    const int v0base = split * (TILES_PER_SPLIT * 16);
    for (int tile = 0; tile < TILES_PER_SPLIT; ++tile) {
        const int v0 = v0base + tile * 16;
        v2f bv[8];
        load_frag8(Who + (size_t)(v0 + ln) * CDIM + ko, bv);
        v8f c = gemm_tile(a, bv);
        const float bias = bho[v0 + ln];
#pragma unroll
        for (int r = 0; r < 8; ++r)
            online_upd(rmax[r], rsum[r], c[r] + bias);
    }

    // merge (max,sum) across the 16 lanes of each half-wave
#pragma unroll
    for (int r = 0; r < 8; ++r) {
        float m = rmax[r], s = rsum[r];
#pragma unroll
        for (int off = 1; off < 16; off <<= 1) {
            const float om = __shfl_xor(m, off, 32);
            const float os = __shfl_xor(s, off, 32);
            online_merge(m, s, om, os);
        }
        rmax[r] = m; rsum[r] = s;
    }

    if (ln == 0) {
#pragma unroll
        for (int r = 0; r < 8; ++r) {
            const int row = m0 + hi * 8 + r;
            pMax[row * NSPLIT + split] = rmax[r];
            pSum[row * NSPLIT + split] = rsum[r];
        }
    }
}

// ---------------------------------------------------------------------------
// Kernel 3: combine the NSPLIT partials per row -> offset = max + log(sumexp)
// ---------------------------------------------------------------------------
__global__ __launch_bounds__(256) void reduce_stats_kernel(
    const float* __restrict__ pMax,
    const float* __restrict__ pSum,
    float* __restrict__ rowOff)      // [4096]
{
    const int row = blockIdx.x * blockDim.x + threadIdx.x;
    if (row >= NROWS) return;
    float m = -INFINITY, s = 0.0f;
#pragma unroll
    for (int sp = 0; sp < NSPLIT; ++sp)
        online_merge(m, s, pMax[row * NSPLIT + sp], pSum[row * NSPLIT + sp]);
    rowOff[row] = m + logf(s);
}

// ---------------------------------------------------------------------------
// Kernel 4: recompute GEMM tiles via WMMA and emit logit - offset.
// ---------------------------------------------------------------------------
__global__ __launch_bounds__(256) void logits_write_kernel(
    const float* __restrict__ C,
    const float* __restrict__ Who,
    const float* __restrict__ bho,
    const float* __restrict__ rowOff,
    float* __restrict__ out)         // [4096, VOCAB]
{
    const int wave  = blockIdx.x * (blockDim.x >> 5) + (threadIdx.x >> 5);
    const int lane  = threadIdx.x & 31;
    const int mtile = wave / NSPLIT;
    const int split = wave % NSPLIT;
    const int m0    = mtile * 16;
    const int ln    = lane & 15;
    const int hi    = lane >> 4;
    const int ko    = hi * 2;

    v2f a[8];
    load_frag8(C + (m0 + ln) * CDIM + ko, a);

    float off[8];
#pragma unroll
    for (int r = 0; r < 8; ++r) off[r] = rowOff[m0 + hi * 8 + r];

    const int v0base = split * (TILES_PER_SPLIT * 16);
    for (int tile = 0; tile < TILES_PER_SPLIT; ++tile) {
        const int v0 = v0base + tile * 16;
        v2f bv[8];
        load_frag8(Who + (size_t)(v0 + ln) * CDIM + ko, bv);
        v8f c = gemm_tile(a, bv);
        const float bias = bho[v0 + ln];
#pragma unroll
        for (int r = 0; r < 8; ++r) {
            const size_t row = (size_t)(m0 + hi * 8 + r);
            out[row * VOCAB + v0 + ln] = c[r] + bias - off[r];
        }
    }
}

// ---------------------------------------------------------------------------
extern "C" void kernel_launch(void* const* d_in, const int* in_sizes, int n_in,
                              void* d_out, int out_size, void* d_ws, size_t ws_size,
                              hipStream_t stream) {
    const int*   tok  = (const int*)  d_in[0];
    const float* h0lr = (const float*)d_in[1];
    const float* h0rl = (const float*)d_in[2];
    const float* emb  = (const float*)d_in[3];
    const float* Wlr  = (const float*)d_in[4];
    const float* blr  = (const float*)d_in[5];
    const float* Wrl  = (const float*)d_in[6];
    const float* brl  = (const float*)d_in[7];
    const float* Who  = (const float*)d_in[8];
    const float* bho  = (const float*)d_in[9];
    float* out = (float*)d_out;

    // workspace layout
    float* C      = (float*)d_ws;                 // 128*32*32   = 131072 floats
    float* pMax   = C + S_LEN * BATCH * CDIM;     // 4096*16     =  65536
    float* pSum   = pMax + NROWS * NSPLIT;        // 4096*16     =  65536
    float* rowOff = pSum + NROWS * NSPLIT;        // 4096

    birnn_scan_kernel<<<1, 1024, 0, stream>>>(tok, h0lr, h0rl, emb,
                                              Wlr, blr, Wrl, brl, C);

    const int nwaves = NMTILE * NSPLIT;           // 4096 waves
    const int wpb = 8;                            // 256-thread blocks
    logits_stats_kernel<<<nwaves / wpb, wpb * 32, 0, stream>>>(C, Who, bho, pMax, pSum);

    reduce_stats_kernel<<<(NROWS + 255) / 256, 256, 0, stream>>>(pMax, pSum, rowOff);

    logits_write_kernel<<<nwaves / wpb, wpb * 32, 0, stream>>>(C, Who, bho, rowOff, out);
}